// Model_29867202576473
// MI455X (gfx1250) — compile-verified
//
#include <hip/hip_runtime.h>
#include <math.h>

// ---------------------------------------------------------------------------
// 2-layer GCN (PyG GCNConv semantics) x2 graphs + column standardization.
// fp32 end-to-end; GEMMs use V_WMMA_F32_16X16X4_F32 (CDNA5 fp32 WMMA).
// ---------------------------------------------------------------------------

typedef float v2f __attribute__((ext_vector_type(2)));
typedef float v8f __attribute__((ext_vector_type(8)));

#define ATOMIC_ADD_F32(p, v) \
  __hip_atomic_fetch_add((p), (v), __ATOMIC_RELAXED, __HIP_MEMORY_SCOPE_AGENT)

static constexpr int DOUT = 128;  // both layer output widths are 128

// ---- small elementwise kernels --------------------------------------------

__global__ void k_fill1(float* __restrict__ p, int n) {
  int i = blockIdx.x * blockDim.x + threadIdx.x;
  if (i < n) p[i] = 1.0f;
}

__global__ void k_deg_accum(float* __restrict__ deg, const int* __restrict__ dst, int E) {
  int e = blockIdx.x * blockDim.x + threadIdx.x;
  if (e < E) ATOMIC_ADD_F32(&deg[dst[e]], 1.0f);
}

__global__ void k_rsqrt_ip(float* __restrict__ p, int n) {
  int i = blockIdx.x * blockDim.x + threadIdx.x;
  if (i < n) p[i] = rsqrtf(p[i]);  // deg >= 1 always (self loop)
}

// out[i,c] = H[i,c] * dinv[i]^2   (the self-loop contribution; also fully
// initializes the accumulation buffer)
__global__ void k_self_loop(const float* __restrict__ H, const float* __restrict__ dinv,
                            float* __restrict__ out, int total) {
  int i = blockIdx.x * blockDim.x + threadIdx.x;
  if (i < total) {
    float d = dinv[i >> 7];
    out[i] = H[i] * d * d;
  }
}

__global__ void k_bias_relu(float* __restrict__ h, const float* __restrict__ b, int total) {
  int i = blockIdx.x * blockDim.x + threadIdx.x;
  if (i < total) h[i] = fmaxf(h[i] + b[i & 127], 0.0f);
}

__global__ void k_bias_add(float* __restrict__ h, const float* __restrict__ b, int total) {
  int i = blockIdx.x * blockDim.x + threadIdx.x;
  if (i < total) h[i] = h[i] + b[i & 127];
}

// ---- edge scatter: out[dst] += H[src] * dinv[src]*dinv[dst] ---------------
// One wave32 per edge; each lane owns 4 contiguous features (float4 gather,
// 4x global_atomic_add_f32 scatter). H and out are L2-resident (51.2 MB each).

__global__ __launch_bounds__(256) void k_scatter(const float* __restrict__ H,
                                                 const float* __restrict__ dinv,
                                                 const int* __restrict__ src,
                                                 const int* __restrict__ dst,
                                                 float* __restrict__ out, int E) {
  int wave = threadIdx.x >> 5;
  int lane = threadIdx.x & 31;
  int e = blockIdx.x * 8 + wave;
  if (e >= E) return;
  int s = src[e];
  int d = dst[e];
  float coef = dinv[s] * dinv[d];
  const float4* hv = (const float4*)(H + (size_t)s * DOUT);
  float4 v = hv[lane];
  float* o = out + (size_t)d * DOUT + lane * 4;
  ATOMIC_ADD_F32(o + 0, v.x * coef);
  ATOMIC_ADD_F32(o + 1, v.y * coef);
  ATOMIC_ADD_F32(o + 2, v.z * coef);
  ATOMIC_ADD_F32(o + 3, v.w * coef);
}

// ---- fp32 WMMA GEMM: C[n,128] = A[n,K] @ W[K,128] -------------------------
// Block = 256 threads = 8 waves; block handles a 16-row stripe, wave w owns
// columns [16w, 16w+16). A tile staged in LDS (padded to kill bank conflicts).
// Fragment layouts per CDNA5 ISA 7.12.2 (32-bit 16x4 A / 4x16 B / 16x16 C).

template <int K>
__global__ __launch_bounds__(256) void k_gemm_wmma(const float* __restrict__ A,
                                                   const float* __restrict__ W,
                                                   float* __restrict__ C, int n) {
  constexpr int LDSP = K + 4;
  __shared__ float As[16 * LDSP];

  const int rowBase = blockIdx.x * 16;
  const int tid = threadIdx.x;

  // Cooperative, coalesced load of the 16 x K A tile.
  for (int idx = tid; idx < 16 * K; idx += 256) {
    int r = idx / K;
    int c = idx - r * K;
    As[r * LDSP + c] = A[(size_t)(rowBase + r) * K + c];
  }
  __syncthreads();

  const int wave = tid >> 5;
  const int lane = tid & 31;
  const int half = lane >> 4;   // 0: lanes 0-15, 1: lanes 16-31
  const int lm   = lane & 15;
  const int colBase = wave * 16;

  v8f acc = {};
  for (int k = 0; k < K; k += 4) {
    const int ka = k + 2 * half;  // lanes 0-15 -> K {k,k+1}; lanes 16-31 -> {k+2,k+3}
    v2f a, b;
    a[0] = As[lm * LDSP + ka];
    a[1] = As[lm * LDSP + ka + 1];
    b[0] = W[(size_t)ka * DOUT + colBase + lm];
    b[1] = W[(size_t)(ka + 1) * DOUT + colBase + lm];
    acc = __builtin_amdgcn_wmma_f32_16x16x4_f32(false, a, false, b, (short)0, acc,
                                                false, false);
  }

  // C/D layout: VGPR g -> row g (lanes 0-15) / row g+8 (lanes 16-31), col = lane%16.
  const int col = colBase + lm;
#pragma unroll
  for (int g = 0; g < 8; ++g) {
    int row = rowBase + g + 8 * half;
    C[(size_t)row * DOUT + col] = acc[g];
  }
}

// ---- column standardization ------------------------------------------------

__global__ void k_zero_stats(float* __restrict__ stats) {
  int i = threadIdx.x;  // 256 threads: [0,128)=sum, [128,256)=sumsq
  stats[i] = 0.0f;
}

__global__ __launch_bounds__(128) void k_col_stats(const float* __restrict__ h,
                                                   float* __restrict__ sum,
                                                   float* __restrict__ sumsq, int n) {
  int c = threadIdx.x;  // 128 columns
  float s = 0.0f, q = 0.0f;
  for (int r = blockIdx.x; r < n; r += gridDim.x) {
    float v = h[(size_t)r * DOUT + c];
    s += v;
    q += v * v;
  }
  ATOMIC_ADD_F32(&sum[c], s);
  ATOMIC_ADD_F32(&sumsq[c], q);
}

__global__ void k_finalize_stats(float* __restrict__ stats, int n) {
  int c = threadIdx.x;  // 128
  float mu = stats[c] / (float)n;
  float var = (stats[128 + c] - (float)n * mu * mu) / (float)(n - 1);  // ddof=1
  stats[256 + c] = mu;
  stats[384 + c] = 1.0f / sqrtf(var);
}

__global__ void k_col_norm(float* __restrict__ h, const float* __restrict__ mu,
                           const float* __restrict__ isd, int total) {
  int i = blockIdx.x * blockDim.x + threadIdx.x;
  if (i < total) {
    int c = i & 127;
    h[i] = (h[i] - mu[c]) * isd[c];
  }
}

// ---------------------------------------------------------------------------

extern "C" void kernel_launch(void* const* d_in, const int* in_sizes, int n_in,
                              void* d_out, int out_size, void* d_ws, size_t ws_size,
                              hipStream_t stream) {
  const float* x1  = (const float*)d_in[0];
  const int*   ei1 = (const int*)d_in[1];
  const float* x2  = (const float*)d_in[2];
  const int*   ei2 = (const int*)d_in[3];
  const float* W0  = (const float*)d_in[4];
  const float* b0  = (const float*)d_in[5];
  const float* W1  = (const float*)d_in[6];
  const float* b1  = (const float*)d_in[7];

  const int IN = 256;
  const int n  = in_sizes[0] / IN;   // 100000
  const int E  = in_sizes[1] / 2;    // 1000000
  const int total = n * DOUT;        // 12.8M

  float* out = (float*)d_out;

  // workspace layout (floats): dinv[n] | bufA[n*128] | bufB[n*128] | stats[512]
  float* ws   = (float*)d_ws;
  float* dinv = ws;
  float* bufA = dinv + n;
  float* bufB = bufA + (size_t)n * DOUT;
  float* stats = bufB + (size_t)n * DOUT;  // sum | sumsq | mu | isd

  const int EW_BLK = 256;
  const int ew_grid = (total + EW_BLK - 1) / EW_BLK;
  const int n_grid  = (n + EW_BLK - 1) / EW_BLK;
  const int e_grid  = (E + EW_BLK - 1) / EW_BLK;
  const int sc_grid = (E + 7) / 8;        // 8 edges (waves) per block
  const int gemm_grid = n / 16;           // 100000 / 16 = 6250

  for (int g = 0; g < 2; ++g) {
    const float* x  = g ? x2 : x1;
    const int*   ei = g ? ei2 : ei1;
    const int* src = ei;       // edge_index[0]
    const int* dst = ei + E;   // edge_index[1]
    float* outg = out + (size_t)g * n * DOUT;

    // degrees (init 1.0 for self loop) -> dinv
    k_fill1<<<n_grid, EW_BLK, 0, stream>>>(dinv, n);
    k_deg_accum<<<e_grid, EW_BLK, 0, stream>>>(dinv, dst, E);
    k_rsqrt_ip<<<n_grid, EW_BLK, 0, stream>>>(dinv, n);

    // layer 1: h = relu(agg(x @ W0) + b0)
    k_gemm_wmma<256><<<gemm_grid, 256, 0, stream>>>(x, W0, bufA, n);
    k_self_loop<<<ew_grid, EW_BLK, 0, stream>>>(bufA, dinv, bufB, total);
    k_scatter<<<sc_grid, 256, 0, stream>>>(bufA, dinv, src, dst, bufB, E);
    k_bias_relu<<<ew_grid, EW_BLK, 0, stream>>>(bufB, b0, total);

    // layer 2: out = agg(h @ W1) + b1   (aggregate straight into d_out slice)
    k_gemm_wmma<128><<<gemm_grid, 256, 0, stream>>>(bufB, W1, bufA, n);
    k_self_loop<<<ew_grid, EW_BLK, 0, stream>>>(bufA, dinv, outg, total);
    k_scatter<<<sc_grid, 256, 0, stream>>>(bufA, dinv, src, dst, outg, E);
    k_bias_add<<<ew_grid, EW_BLK, 0, stream>>>(outg, b1, total);

    // column standardization (ddof=1)
    k_zero_stats<<<1, 256, 0, stream>>>(stats);
    k_col_stats<<<1024, 128, 0, stream>>>(outg, stats, stats + 128, n);
    k_finalize_stats<<<1, 128, 0, stream>>>(stats, n);
    k_col_norm<<<ew_grid, EW_BLK, 0, stream>>>(outg, stats + 256, stats + 384, total);
  }
}